// VGAE_encoder_33663953666491
// MI455X (gfx1250) — compile-verified
//
#include <hip/hip_runtime.h>
#include <hip/hip_bf16.h>

typedef float v2f __attribute__((ext_vector_type(2)));
typedef float v8f __attribute__((ext_vector_type(8)));

#define FDIM 256

// ---------------- degree / normalization ----------------
__global__ void deg_init_kernel(float* deg, int n) {
    int i = blockIdx.x * blockDim.x + threadIdx.x;
    if (i < n) deg[i] = 1.0f;  // self-loop
}

__global__ void deg_count_kernel(const int* __restrict__ dst, float* deg, int n_edges) {
    int i = blockIdx.x * blockDim.x + threadIdx.x;
    if (i < n_edges) atomicAdd(&deg[dst[i]], 1.0f);
}

__global__ void dis_kernel(float* deg_dis, int n) {
    int i = blockIdx.x * blockDim.x + threadIdx.x;
    if (i < n) deg_dis[i] = rsqrtf(deg_dis[i]);  // deg >= 1 always
}

// ---------------- weight/bias concat for fused mu|logvar ----------------
__global__ void prep_cat_kernel(const float* __restrict__ Wmu, const float* __restrict__ Wlv,
                                const float* __restrict__ bmu, const float* __restrict__ blv,
                                float* __restrict__ Wcat, float* __restrict__ bcat) {
    int t = blockIdx.x * blockDim.x + threadIdx.x;  // 256*256 threads
    if (t >= 256 * 256) return;
    int k = t >> 8, n = t & 255;
    Wcat[t] = (n < 128) ? Wmu[k * 128 + n] : Wlv[k * 128 + (n - 128)];
    if (t < 256) bcat[t] = (t < 128) ? bmu[t] : blv[t - 128];
}

// ---------------- exact-f32 WMMA GEMM: C[M,Nc] = act(A)[M,K] @ B[K,Nc] ----------------
// One wave computes a 16(M) x 64(N) tile with V_WMMA_F32_16X16X4_F32 (K stepped by 4).
// All 4 B fragments of a K-step are loaded into distinct registers BEFORE the 4 WMMAs,
// so loads pipeline instead of serializing each WMMA behind s_wait_loadcnt 0.
// A 16x4 frag: lanes 0-15 -> (K=k0,k0+1), lanes 16-31 -> (K=k0+2,k0+3), row = m0+(lane&15).
// B 4x16 frag: mirrored. C/D: VGPR j -> row m0 + 8*(lane>=16) + j, col n0 + (lane&15).
template <bool RELU>
__global__ void gemm16x64_wmma_f32(const float* __restrict__ A, const float* __restrict__ B,
                                   float* __restrict__ C, int M, int K, int Nc) {
    const int lane = threadIdx.x & 31;
    const int half = lane >> 4;
    const int l15  = lane & 15;
    const int n0 = blockIdx.x * 64;
    const int m0 = blockIdx.y * 16;

    v8f acc[4] = {{}, {}, {}, {}};
    const float* arow = A + (size_t)(m0 + l15) * (size_t)K;

    #pragma unroll 4
    for (int k0 = 0; k0 < K; k0 += 4) {
        float a0 = arow[k0 + 2 * half];
        float a1 = arow[k0 + 2 * half + 1];
        if (RELU) { a0 = fmaxf(a0, 0.0f); a1 = fmaxf(a1, 0.0f); }
        v2f a; a.x = a0; a.y = a1;

        const float* brow0 = B + (size_t)(k0 + 2 * half) * (size_t)Nc + n0 + l15;
        v2f b[4];
        #pragma unroll
        for (int t = 0; t < 4; ++t) {
            b[t].x = brow0[16 * t];
            b[t].y = brow0[Nc + 16 * t];
        }
        #pragma unroll
        for (int t = 0; t < 4; ++t)
            acc[t] = __builtin_amdgcn_wmma_f32_16x16x4_f32(false, a, false, b[t],
                                                           (short)0, acc[t], false, false);
    }

    float* crow = C + (size_t)(m0 + half * 8) * (size_t)Nc + n0 + l15;
    #pragma unroll
    for (int t = 0; t < 4; ++t)
        #pragma unroll
        for (int j = 0; j < 8; ++j)
            crow[(size_t)j * (size_t)Nc + 16 * t] = acc[t][j];
}

// ---------------- aggregation init: out = bias + deg^-1 * h (self loop) ----------------
// Split output: features [0,128) -> outA (+ n*strideOut), [128,256) -> outB.
__global__ void agg_init_kernel(const float* __restrict__ H, const float* __restrict__ dis,
                                const float* __restrict__ bias,
                                float* __restrict__ outA, float* __restrict__ outB,
                                int strideOut, int n_nodes) {
    int t = blockIdx.x * blockDim.x + threadIdx.x;  // n_nodes * 64 (float4 per thread)
    if (t >= n_nodes * 64) return;
    int n = t >> 6;
    int f = (t & 63) * 4;  // never straddles the 128 boundary
    float w = dis[n]; w = w * w;  // dis[n]^2 == 1/deg[n]
    float4 h  = *(const float4*)(H + (size_t)n * FDIM + f);
    float4 bb = *(const float4*)(bias + f);
    float4 r;
    r.x = bb.x + w * h.x; r.y = bb.y + w * h.y;
    r.z = bb.z + w * h.z; r.w = bb.w + w * h.w;
    float* op = ((f < 128) ? outA : outB) + (size_t)n * strideOut + (f & 127);
    *(float4*)op = r;
}

// ---------------- edge scatter-add: out[dst] += dis[src]*dis[dst] * h[src] ----------------
// One wave32 per edge; lane l covers features [8l, 8l+8). Edge state forced scalar with
// readfirstlane (e is wave-uniform) so address/norm math lands on SALU.
__global__ void agg_edge_kernel(const int* __restrict__ src, const int* __restrict__ dst,
                                const float* __restrict__ dis, const float* __restrict__ H,
                                float* __restrict__ outA, float* __restrict__ outB,
                                int strideOut, int n_edges) {
    int e = blockIdx.x * (blockDim.x >> 5) + (threadIdx.x >> 5);
    if (e >= n_edges) return;
    int lane = threadIdx.x & 31;
    int s = __builtin_amdgcn_readfirstlane(src[e]);
    int d = __builtin_amdgcn_readfirstlane(dst[e]);
    float norm = dis[s] * dis[d];
    int f0 = lane * 8;
    const float4* hp = (const float4*)(H + (size_t)s * FDIM + f0);
    float4 h0 = hp[0];
    float4 h1 = hp[1];
    float* ob = ((f0 < 128) ? outA : outB) + (size_t)d * strideOut + (f0 & 127);
    atomicAdd(ob + 0, norm * h0.x);
    atomicAdd(ob + 1, norm * h0.y);
    atomicAdd(ob + 2, norm * h0.z);
    atomicAdd(ob + 3, norm * h0.w);
    atomicAdd(ob + 4, norm * h1.x);
    atomicAdd(ob + 5, norm * h1.y);
    atomicAdd(ob + 6, norm * h1.z);
    atomicAdd(ob + 7, norm * h1.w);
}

extern "C" void kernel_launch(void* const* d_in, const int* in_sizes, int n_in,
                              void* d_out, int out_size, void* d_ws, size_t ws_size,
                              hipStream_t stream) {
    const float* x   = (const float*)d_in[0];
    const int*   ei  = (const int*)  d_in[1];
    const float* W1  = (const float*)d_in[2];
    const float* b1  = (const float*)d_in[3];
    const float* Wmu = (const float*)d_in[4];
    const float* bmu = (const float*)d_in[5];
    const float* Wlv = (const float*)d_in[6];
    const float* blv = (const float*)d_in[7];
    float* out = (float*)d_out;

    const int N = in_sizes[0] / FDIM;   // 100000 (divisible by 16)
    const int E = in_sizes[1] / 2;      // 1600000
    const int* srcI = ei;
    const int* dstI = ei + E;

    // workspace layout (floats)
    float* f_dis  = (float*)d_ws;                  // N
    float* f_bcat = f_dis  + N;                    // 256
    float* f_wcat = f_bcat + 256;                  // 256*256
    float* f_HZ   = f_wcat + 256 * 256;            // N*256 (H, later reused as Z)
    float* f_O1   = f_HZ   + (size_t)N * FDIM;     // N*256

    const int T = 256;
    dim3 gemmGrid(FDIM / 64, N / 16);

    // normalization: deg -> dis = rsqrt(deg)
    deg_init_kernel<<<(N + T - 1) / T, T, 0, stream>>>(f_dis, N);
    deg_count_kernel<<<(E + T - 1) / T, T, 0, stream>>>(dstI, f_dis, E);
    dis_kernel<<<(N + T - 1) / T, T, 0, stream>>>(f_dis, N);

    // fused [W_mu | W_lv], [b_mu | b_lv]
    prep_cat_kernel<<<(256 * 256) / T, T, 0, stream>>>(Wmu, Wlv, bmu, blv, f_wcat, f_bcat);

    // ---- layer 1: H = X @ W1 ; O1 = b1 + selfloop + scatter-add ----
    gemm16x64_wmma_f32<false><<<gemmGrid, 32, 0, stream>>>(x, W1, f_HZ, N, FDIM, FDIM);
    agg_init_kernel<<<((size_t)N * 64 + T - 1) / T, T, 0, stream>>>(
        f_HZ, f_dis, b1, f_O1, f_O1 + 128, FDIM, N);
    agg_edge_kernel<<<((size_t)E * 32 + T - 1) / T, T, 0, stream>>>(
        srcI, dstI, f_dis, f_HZ, f_O1, f_O1 + 128, FDIM, E);

    // ---- layer 2 (mu & logvar fused): Z = relu(O1) @ Wcat ; out = agg(Z) ----
    gemm16x64_wmma_f32<true><<<gemmGrid, 32, 0, stream>>>(f_O1, f_wcat, f_HZ, N, FDIM, FDIM);
    agg_init_kernel<<<((size_t)N * 64 + T - 1) / T, T, 0, stream>>>(
        f_HZ, f_dis, f_bcat, out, out + (size_t)N * 128, 128, N);
    agg_edge_kernel<<<((size_t)E * 32 + T - 1) / T, T, 0, stream>>>(
        srcI, dstI, f_dis, f_HZ, out, out + (size_t)N * 128, 128, E);
}